// BahdanauDecoder_10299331576043
// MI455X (gfx1250) — compile-verified
//
#include <hip/hip_runtime.h>
#include <math.h>

// Problem sizes (compile-time)
static constexpr int kH  = 1024;   // hidden
static constexpr int kH2 = 2048;   // 2*H
static constexpr int kH3 = 3072;   // 3*H
static constexpr int kS  = 2048;   // encoder length
static constexpr int kV  = 50257;  // vocab
static constexpr int kTiles = (kV + 15) / 16;  // 3142 WMMA row-tiles

// Workspace layout (float offsets)
static constexpr int WS_W2      = 0;          // 1024  : attn_W[:,H:]^T @ other
static constexpr int WS_SCORES  = 1024;       // 2048
static constexpr int WS_ATTN    = 3072;       // 2048
static constexpr int WS_CTXPART = 5120;       // 8*1024 partial context sums
static constexpr int WS_X       = 13312;      // 2048  : [emb, context]
static constexpr int WS_Y       = 15360;      // 2048  : [h_new, context]
static constexpr int WS_GX      = 17408;      // 3072
static constexpr int WS_GH      = 20480;      // 3072
static constexpr int WS_RED     = 23552;      // 2     : {max, log(sumexp)}
static constexpr int WS_LOGITS  = 23560;      // 50257

typedef float v2f __attribute__((ext_vector_type(2)));
typedef float v8f __attribute__((ext_vector_type(8)));

// ---- Stage 0: w2[k] = sum_h other[h] * attn_W[h][H+k]  (coalesced column walk)
__global__ void k_w2(const float* __restrict__ attn_W, const float* __restrict__ other,
                     float* __restrict__ ws) {
  int k = blockIdx.x * 256 + threadIdx.x;
  if (k >= kH) return;
  float acc = 0.f;
  for (int h = 0; h < kH; ++h)
    acc += other[h] * attn_W[(size_t)h * kH2 + kH + k];
  ws[WS_W2 + k] = acc;
}

// ---- Stage 1: scores[s] = enc[s] . w2   (one wave32 per row, shfl reduce)
__global__ void k_scores(const float* __restrict__ enc, float* __restrict__ ws) {
  int wave = (blockIdx.x * blockDim.x + threadIdx.x) >> 5;
  int lane = threadIdx.x & 31;
  if (wave >= kS) return;
  const float* row = enc + (size_t)wave * kH;
  const float* w2  = ws + WS_W2;
  float acc = 0.f;
  for (int j = lane; j < kH; j += 32) acc += row[j] * w2[j];
  for (int o = 16; o > 0; o >>= 1) acc += __shfl_down(acc, o, 32);
  if (lane == 0) ws[WS_SCORES + wave] = acc;
}

// ---- Stage 2: softmax over S (single block -> deterministic), write attn out
__global__ void k_softmax(float* __restrict__ ws, float* __restrict__ out) {
  __shared__ float sd[256];
  int tid = threadIdx.x;
  float mx = -INFINITY;
  for (int i = tid; i < kS; i += 256) mx = fmaxf(mx, ws[WS_SCORES + i]);
  sd[tid] = mx; __syncthreads();
  for (int o = 128; o > 0; o >>= 1) { if (tid < o) sd[tid] = fmaxf(sd[tid], sd[tid + o]); __syncthreads(); }
  mx = sd[0]; __syncthreads();
  float sum = 0.f;
  for (int i = tid; i < kS; i += 256) sum += expf(ws[WS_SCORES + i] - mx);
  sd[tid] = sum; __syncthreads();
  for (int o = 128; o > 0; o >>= 1) { if (tid < o) sd[tid] += sd[tid + o]; __syncthreads(); }
  float inv = 1.0f / sd[0];
  for (int i = tid; i < kS; i += 256) {
    float a = expf(ws[WS_SCORES + i] - mx) * inv;
    ws[WS_ATTN + i] = a;
    out[kV + kH + i] = a;               // attn_weights output
  }
}

// ---- Stage 3: partial context: block = (colgroup 0..3) x (s-chunk 0..7)
__global__ void k_ctx_partial(const float* __restrict__ enc, float* __restrict__ ws) {
  int cg  = blockIdx.x & 3;
  int sc  = blockIdx.x >> 2;
  int col = cg * 256 + threadIdx.x;
  const float* attn = ws + WS_ATTN;
  float acc = 0.f;
  int s0 = sc * 256;
  for (int s = s0; s < s0 + 256; ++s)
    acc += attn[s] * enc[(size_t)s * kH + col];
  ws[WS_CTXPART + sc * kH + col] = acc;
}

// ---- Stage 4: combine context; build x=[emb,ctx]; y[H:]=ctx
__global__ void k_ctx_combine(const float* __restrict__ embedding, const int* __restrict__ word,
                              float* __restrict__ ws) {
  int col = blockIdx.x * 256 + threadIdx.x;
  if (col >= kH) return;
  float c = 0.f;
  for (int q = 0; q < 8; ++q) c += ws[WS_CTXPART + q * kH + col];
  ws[WS_X + kH + col] = c;
  ws[WS_Y + kH + col] = c;
  ws[WS_X + col] = embedding[(size_t)word[0] * kH + col];
}

// ---- Stage 5: GRU GEMVs: one wave per row r of W_ih (2048) and W_hh (1024)
__global__ void k_gru_gemv(const float* __restrict__ W_ih, const float* __restrict__ W_hh,
                           const float* __restrict__ b_ih, const float* __restrict__ b_hh,
                           const float* __restrict__ hvec, float* __restrict__ ws) {
  int wave = (blockIdx.x * blockDim.x + threadIdx.x) >> 5;
  int lane = threadIdx.x & 31;
  if (wave >= kH3) return;
  const float* x = ws + WS_X;
  const float* wrow = W_ih + (size_t)wave * kH2;
  const float* hrow = W_hh + (size_t)wave * kH;
  float ax = 0.f, ah = 0.f;
  for (int j = lane; j < kH2; j += 32) ax += wrow[j] * x[j];
  for (int j = lane; j < kH;  j += 32) ah += hrow[j] * hvec[j];
  for (int o = 16; o > 0; o >>= 1) {
    ax += __shfl_down(ax, o, 32);
    ah += __shfl_down(ah, o, 32);
  }
  if (lane == 0) {
    ws[WS_GX + wave] = ax + b_ih[wave];
    ws[WS_GH + wave] = ah + b_hh[wave];
  }
}

// ---- Stage 6: gates + h_new ; y[0:H] = h_new
__global__ void k_gates(const float* __restrict__ hvec, float* __restrict__ ws,
                        float* __restrict__ out) {
  int i = blockIdx.x * 256 + threadIdx.x;
  if (i >= kH) return;
  const float* gx = ws + WS_GX;
  const float* gh = ws + WS_GH;
  float r  = 1.f / (1.f + expf(-(gx[i] + gh[i])));
  float z  = 1.f / (1.f + expf(-(gx[kH + i] + gh[kH + i])));
  float n  = tanhf(gx[2 * kH + i] + r * gh[2 * kH + i]);
  float hn = (1.f - z) * n + z * hvec[i];
  ws[WS_Y + i] = hn;
  out[kV + i] = hn;                     // hidden output
}

// ---- Stage 7: logits = out_W @ y + out_b via V_WMMA_F32_16X16X4_F32.
// One wave owns 16 rows. y-chunk replicated into every B column, so every
// column of D carries the 16 row-dot-products; lanes 0/16 extract them.
__global__ void k_logits_wmma(const float* __restrict__ outW, const float* __restrict__ outb,
                              float* __restrict__ ws) {
  __shared__ float sy[kH2];             // stage y once per block (ds path)
  for (int i = threadIdx.x; i < kH2; i += 256) sy[i] = ws[WS_Y + i];
  __syncthreads();

  int wave = (blockIdx.x * blockDim.x + threadIdx.x) >> 5;
  int lane = threadIdx.x & 31;
  if (wave < kTiles) {
    int rowBase = wave * 16;
    int m   = lane & 15;
    int row = rowBase + m;
    if (row > kV - 1) row = kV - 1;     // clamp tail-tile loads in-bounds
    int koff = (lane < 16) ? 0 : 2;     // A/B 32-bit layout: v0:{K0|K2}, v1:{K1|K3}
    const float* A = outW + (size_t)row * kH2 + koff;
    const float* y = sy + koff;
    v8f acc0 = {0.f, 0.f, 0.f, 0.f, 0.f, 0.f, 0.f, 0.f};
    v8f acc1 = {0.f, 0.f, 0.f, 0.f, 0.f, 0.f, 0.f, 0.f};
    #pragma unroll 4
    for (int k = 0; k < kH2; k += 8) {
      v2f a0 = *(const v2f*)(A + k);
      v2f b0 = *(const v2f*)(y + k);
      v2f a1 = *(const v2f*)(A + k + 4);
      v2f b1 = *(const v2f*)(y + k + 4);
      acc0 = __builtin_amdgcn_wmma_f32_16x16x4_f32(false, a0, false, b0, (short)0, acc0, false, false);
      acc1 = __builtin_amdgcn_wmma_f32_16x16x4_f32(false, a1, false, b1, (short)0, acc1, false, false);
    }
    v8f acc = acc0 + acc1;
    if (m == 0) {                       // lane 0 -> M=0..7, lane 16 -> M=8..15
      int half = lane >> 4;
      #pragma unroll
      for (int vv = 0; vv < 8; ++vv) {
        int r = rowBase + half * 8 + vv;
        if (r < kV) ws[WS_LOGITS + r] = acc[vv] + outb[r];
      }
    }
  }
}

// ---- Stage 8: single-block max / log-sum-exp over V (deterministic)
__global__ void k_lse(float* __restrict__ ws) {
  __shared__ float sd[1024];
  int tid = threadIdx.x;
  const float* lg = ws + WS_LOGITS;
  float mx = -INFINITY;
  for (int i = tid; i < kV; i += 1024) mx = fmaxf(mx, lg[i]);
  sd[tid] = mx; __syncthreads();
  for (int o = 512; o > 0; o >>= 1) { if (tid < o) sd[tid] = fmaxf(sd[tid], sd[tid + o]); __syncthreads(); }
  mx = sd[0]; __syncthreads();
  float sum = 0.f;
  for (int i = tid; i < kV; i += 1024) sum += expf(lg[i] - mx);
  sd[tid] = sum; __syncthreads();
  for (int o = 512; o > 0; o >>= 1) { if (tid < o) sd[tid] += sd[tid + o]; __syncthreads(); }
  if (tid == 0) { ws[WS_RED] = mx; ws[WS_RED + 1] = logf(sd[0]); }
}

// ---- Stage 9: log_probs = logits - max - log(sumexp)
__global__ void k_writeout(const float* __restrict__ ws, float* __restrict__ out) {
  int i = blockIdx.x * 256 + threadIdx.x;
  if (i >= kV) return;
  out[i] = ws[WS_LOGITS + i] - ws[WS_RED] - ws[WS_RED + 1];
}

extern "C" void kernel_launch(void* const* d_in, const int* in_sizes, int n_in,
                              void* d_out, int out_size, void* d_ws, size_t ws_size,
                              hipStream_t stream) {
  const int*   word   = (const int*)  d_in[0];   // last_output_word (1,)
  const float* hvec   = (const float*)d_in[1];   // last_hidden (1,1,H)
  const float* enc    = (const float*)d_in[2];   // encoder_outputs (S,1,H)
  const float* emb    = (const float*)d_in[3];   // embedding (V,H)
  const float* attn_W = (const float*)d_in[4];   // (H, 2H)
  // d_in[5] = attn_b : uniform shift of scores -> cancels under softmax
  const float* other  = (const float*)d_in[6];   // (1,H)
  const float* W_ih   = (const float*)d_in[7];   // (3H, 2H)
  const float* W_hh   = (const float*)d_in[8];   // (3H, H)
  const float* b_ih   = (const float*)d_in[9];
  const float* b_hh   = (const float*)d_in[10];
  const float* out_W  = (const float*)d_in[11];  // (V, 2H)
  const float* out_b  = (const float*)d_in[12];
  float* out = (float*)d_out;                    // [log_probs V | hidden H | attn S]
  float* ws  = (float*)d_ws;

  k_w2         <<<4,                 256, 0, stream>>>(attn_W, other, ws);
  k_scores     <<<kS / 8,            256, 0, stream>>>(enc, ws);
  k_softmax    <<<1,                 256, 0, stream>>>(ws, out);
  k_ctx_partial<<<32,                256, 0, stream>>>(enc, ws);
  k_ctx_combine<<<4,                 256, 0, stream>>>(emb, word, ws);
  k_gru_gemv   <<<kH3 / 8,           256, 0, stream>>>(W_ih, W_hh, b_ih, b_hh, hvec, ws);
  k_gates      <<<4,                 256, 0, stream>>>(hvec, ws, out);
  k_logits_wmma<<<(kTiles + 7) / 8,  256, 0, stream>>>(out_W, out_b, ws);
  k_lse        <<<1,                1024, 0, stream>>>(ws);
  k_writeout   <<<(kV + 255) / 256,  256, 0, stream>>>(ws, out);
}